// OptimizedDNN_37160057045546
// MI455X (gfx1250) — compile-verified
//
#include <hip/hip_runtime.h>
#include <hip/hip_bf16.h>

#define VOCAB 100000
#define DIM   512
#define BATCH 4096
#define SEQ   200
#define HID   256
#define NOUT  20
#define K2D   (2 * DIM)   // 1024

typedef float v2f __attribute__((ext_vector_type(2)));
typedef float v8f __attribute__((ext_vector_type(8)));

// ---------------------------------------------------------------------------
// Phase 1: embedding gather + mean/max pooling.
// One block per batch row (4096 blocks, 256 threads = 8 wave32).
// Thread t owns columns [2t, 2t+1]; streams SEQ rows, accumulating sum & max
// in registers. Indices staged once in LDS. Prefetch 16 rows ahead.
// Output rep[b][0:512] = mean, rep[b][512:1024] = max.
// ---------------------------------------------------------------------------
__global__ __launch_bounds__(256) void pool_kernel(const float* __restrict__ emb,
                                                   const int* __restrict__ x,
                                                   const int* __restrict__ lengths,
                                                   float* __restrict__ rep) {
    __shared__ int s_idx[SEQ];
    const int b = blockIdx.x;
    const int t = threadIdx.x;
    if (t < SEQ) s_idx[t] = x[b * SEQ + t];
    __syncthreads();

    const int c = 2 * t;  // column pair owned by this thread
    float sx = 0.f, sy = 0.f;
    float mx = -INFINITY, my = -INFINITY;

    #pragma unroll 4
    for (int s = 0; s < SEQ; ++s) {
        if (((t & 15) == 0) && (s + 16 < SEQ)) {
            // one lane per 128B line prefetches the row 16 iterations ahead
            __builtin_prefetch(emb + (size_t)s_idx[s + 16] * DIM + c, 0, 0);
        }
        const float2 v = *(const float2*)(emb + (size_t)s_idx[s] * DIM + c);
        sx += v.x; sy += v.y;
        mx = fmaxf(mx, v.x);
        my = fmaxf(my, v.y);
    }

    const float invlen = 1.0f / (float)lengths[b];
    float* outb = rep + (size_t)b * K2D;
    *(float2*)(outb + c)       = make_float2(sx * invlen, sy * invlen);
    *(float2*)(outb + DIM + c) = make_float2(mx, my);
}

// ---------------------------------------------------------------------------
// Phase 2: h = relu(rep @ W1 + b1) via V_WMMA_F32_16X16X4_F32 (f32-exact).
// One wave per 16x16 output tile; 256 K-steps of K=4 over K2D=1024.
// A fragment (16x4 f32): lane L holds A[M=L%16][K=base+(L/16)*2 + {0,1}].
// B fragment (4x16 f32): lane L holds B[K=base+(L/16)*2 + {0,1}][N=L%16].
// C/D (16x16 f32, 8 VGPRs): VGPR j of lane L -> row m0 + j + (L>=16 ? 8 : 0),
// col n0 + L%16.
// ---------------------------------------------------------------------------
__global__ __launch_bounds__(256) void mlp1_wmma_kernel(const float* __restrict__ rep,
                                                        const float* __restrict__ W1,
                                                        const float* __restrict__ b1,
                                                        float* __restrict__ h) {
    const int wave = threadIdx.x >> 5;
    const int lane = threadIdx.x & 31;
    const int tile = blockIdx.x * 8 + wave;   // 4096 tiles total
    const int mt = tile >> 4;                 // HID/16 = 16 N-tiles
    const int nt = tile & 15;
    const int m0 = mt * 16;
    const int n0 = nt * 16;
    const int lm = lane & 15;
    const int lk = (lane >> 4) * 2;

    v8f c = {};
    const float* arow = rep + (size_t)(m0 + lm) * K2D + lk;     // A row, K-offset lk
    const float* bcol = W1 + (size_t)lk * HID + (n0 + lm);      // B col, K-offset lk

    for (int k = 0; k < K2D; k += 4) {
        const v2f a = *(const v2f*)(arow + k);                  // A[m][k+lk], A[m][k+lk+1]
        v2f bm;
        bm.x = bcol[(size_t)k * HID];                           // B[k+lk  ][n]
        bm.y = bcol[(size_t)(k + 1) * HID];                     // B[k+lk+1][n]
        c = __builtin_amdgcn_wmma_f32_16x16x4_f32(
                /*neg_a=*/false, a, /*neg_b=*/false, bm,
                /*c_mod=*/(short)0, c, /*reuse_a=*/false, /*reuse_b=*/false);
    }

    const int col  = n0 + lm;
    const float bias = b1[col];
    const int rbase = m0 + ((lane >> 4) << 3);
    #pragma unroll
    for (int j = 0; j < 8; ++j) {
        float v = c[j] + bias;
        h[(size_t)(rbase + j) * HID + col] = v > 0.f ? v : 0.f;
    }
}

// ---------------------------------------------------------------------------
// Phase 3: logits = h @ W2 + b2. N=20 is too skinny for WMMA to pay off
// (42 MFLOP total); W2 staged in LDS, 20 f32 accumulators per thread,
// one batch row per thread.
// ---------------------------------------------------------------------------
__global__ __launch_bounds__(256) void mlp2_kernel(const float* __restrict__ h,
                                                   const float* __restrict__ W2,
                                                   const float* __restrict__ b2,
                                                   float* __restrict__ out) {
    __shared__ float s_w[HID * NOUT];
    __shared__ float s_b[NOUT];
    const int t = threadIdx.x;
    for (int i = t; i < HID * NOUT; i += 256) s_w[i] = W2[i];
    if (t < NOUT) s_b[t] = b2[t];
    __syncthreads();

    const int b = blockIdx.x * 256 + t;
    float acc[NOUT];
    #pragma unroll
    for (int o = 0; o < NOUT; ++o) acc[o] = s_b[o];

    const float* hrow = h + (size_t)b * HID;
    for (int k = 0; k < HID; ++k) {
        const float hv = hrow[k];
        #pragma unroll
        for (int o = 0; o < NOUT; ++o) acc[o] = fmaf(hv, s_w[k * NOUT + o], acc[o]);
    }

    float* orow = out + (size_t)b * NOUT;
    #pragma unroll
    for (int o = 0; o < NOUT; ++o) orow[o] = acc[o];
}

// ---------------------------------------------------------------------------
// Launch wrapper. Input order (setup_inputs dict):
//   0 emb_table [VOCAB*DIM] f32     1 W1 [1024*256] f32   2 b1 [256] f32
//   3 W2 [256*20] f32               4 b2 [20] f32
//   5 x  [4096*200] int             6 lengths [4096] int  7 bows [4096] f32
// Output: logits [4096*20] f32.
// Workspace: rep (16 MB) + h (4 MB) in d_ws.
// ---------------------------------------------------------------------------
extern "C" void kernel_launch(void* const* d_in, const int* in_sizes, int n_in,
                              void* d_out, int out_size, void* d_ws, size_t ws_size,
                              hipStream_t stream) {
    (void)in_sizes; (void)n_in; (void)out_size; (void)ws_size;

    const float* emb     = (const float*)d_in[0];
    const float* W1      = (const float*)d_in[1];
    const float* b1      = (const float*)d_in[2];
    const float* W2      = (const float*)d_in[3];
    const float* b2      = (const float*)d_in[4];
    const int*   x       = (const int*)d_in[5];
    const int*   lengths = (const int*)d_in[6];
    float*       logits  = (float*)d_out;

    float* rep = (float*)d_ws;                       // [BATCH, K2D]
    float* h   = rep + (size_t)BATCH * K2D;          // [BATCH, HID]

    pool_kernel<<<BATCH, 256, 0, stream>>>(emb, x, lengths, rep);

    const int tiles  = (BATCH / 16) * (HID / 16);    // 4096
    const int blocks = tiles / 8;                    // 8 waves per block
    mlp1_wmma_kernel<<<blocks, 256, 0, stream>>>(rep, W1, b1, h);

    mlp2_kernel<<<BATCH / 256, 256, 0, stream>>>(h, W2, b2, logits);
}